// MpnnLayer_57140244906224
// MI455X (gfx1250) — compile-verified
//
#include <hip/hip_runtime.h>
#include <math.h>

// MPNN layer for MI455X / gfx1250 (wave32, WMMA f32 16x16x4).
// B=8, N=256, F=64, E=16, M=64, NUM_EDGE_TYPES=4

#define B_ 8
#define N_ 256
#define F_ 64
#define E_ 16
#define M_ 64

#if defined(__gfx1250__) && __has_builtin(__builtin_amdgcn_global_load_async_to_lds_b128)
#define USE_ASYNC_LDS 1
#else
#define USE_ASYNC_LDS 0
#endif

typedef float v2f __attribute__((ext_vector_type(2)));
typedef float v8f __attribute__((ext_vector_type(8)));
typedef int   v4i_t __attribute__((__vector_size__(4 * sizeof(int))));

// D = A(16x4 f32) x B(4x16 f32) + C(16x16 f32), wave32.
static __device__ __forceinline__ v8f wmma4(v2f a, v2f b, v8f c) {
  return __builtin_amdgcn_wmma_f32_16x16x4_f32(
      /*neg_a=*/false, a, /*neg_b=*/false, b,
      /*c_mod=*/(short)0, c, /*reuse_a=*/false, /*reuse_b=*/false);
}

// ---------------------------------------------------------------------------
// K1: mi = nodes @ Wm[0:64,:], mj = nodes @ Wm[64:128,:]
// grid: 128 blocks (b * 16 row-tiles), 256 threads = 8 waves.
// waves 0-3 -> mi n-tiles, waves 4-7 -> mj n-tiles. K = 64 (16 WMMA steps).
// ---------------------------------------------------------------------------
__global__ void k1_project(const float* __restrict__ nodes,
                           const float* __restrict__ Wm,
                           float* __restrict__ mi,
                           float* __restrict__ mj) {
  const int blk  = blockIdx.x;
  const int b    = blk >> 4;
  const int i0   = (blk & 15) << 4;
  const int tid  = threadIdx.x;
  const int lane = tid & 31;
  const int w    = tid >> 5;
  const int mat  = w >> 2;           // 0 = mi (W_self), 1 = mj (W_other)
  const int n0   = (w & 3) << 4;
  const int sel  = (lane < 16) ? 0 : 2;
  const int col  = lane & 15;
  const int roff = (lane < 16) ? 0 : 8;

  const float* Arow = nodes + (size_t)(b * N_ + i0 + col) * F_;
  const float* Bmat = Wm + (size_t)(mat * 64) * M_;

  v8f c = {};
#pragma unroll
  for (int k = 0; k < F_; k += 4) {
    v2f a, bb;
    a.x  = Arow[k + sel];
    a.y  = Arow[k + 1 + sel];
    bb.x = Bmat[(k + sel) * M_ + n0 + col];
    bb.y = Bmat[(k + 1 + sel) * M_ + n0 + col];
    c = wmma4(a, bb, c);
  }

  float* dst = mat ? mj : mi;
#pragma unroll
  for (int r = 0; r < 8; ++r) {
    const int gr = b * N_ + i0 + r + roff;
    dst[(size_t)gr * M_ + n0 + col] = c[r];
  }
}

// ---------------------------------------------------------------------------
// K2: stream edges ONCE into LDS (async global->LDS when available), then
// produce mask (f32), deg, esum[b,i,e] = sum_j mask*edges from LDS.
// grid: B*N = 2048 blocks, 256 threads. Bandwidth-critical pass (~33.5 MB).
// ---------------------------------------------------------------------------
__global__ void k2_mask_esum(const float* __restrict__ edges,
                             float* __restrict__ mask,
                             float* __restrict__ deg,
                             float* __restrict__ esum) {
  __shared__ float s_edges[N_ * E_];    // 16 KB: full (b,i) edge row [j][e]
  __shared__ float s_mask[N_];
  __shared__ float s_part[256];

  const int bid = blockIdx.x;           // b*N + i
  const int t   = threadIdx.x;
  const size_t base = (size_t)bid * N_ * E_;

  // stage the full 16KB edge row in LDS: 1024 x b128, 4 per thread
#if USE_ASYNC_LDS
#pragma unroll
  for (int it = 0; it < 4; ++it) {
    const int idx = (t + (it << 8)) << 2;     // float offset, 16B aligned
    __builtin_amdgcn_global_load_async_to_lds_b128(
        (__attribute__((address_space(1))) v4i_t*)(edges + base + idx),
        (__attribute__((address_space(3))) v4i_t*)(s_edges + idx), 0, 0);
  }
#if __has_builtin(__builtin_amdgcn_s_wait_asynccnt)
  __builtin_amdgcn_s_wait_asynccnt(0);
#else
  asm volatile("s_wait_asynccnt 0" ::: "memory");
#endif
#else
#pragma unroll
  for (int it = 0; it < 4; ++it) {
    const int idx = (t + (it << 8)) << 2;
    *(float4*)(s_edges + idx) = *(const float4*)(edges + base + idx);
  }
#endif
  __syncthreads();

  // per-j mask from the 4 edge-type channels (read back from LDS)
  {
    const float4 et = *(const float4*)(s_edges + t * E_);
    const float m = (et.x != 0.f || et.y != 0.f || et.z != 0.f || et.w != 0.f)
                        ? 1.f : 0.f;
    s_mask[t] = m;
    mask[(size_t)bid * N_ + t] = m;
  }
  __syncthreads();

  // esum[e] = sum_j edges[j,e] * mask[j]; 16 lanes per e-column, all from LDS
  {
    const int e  = t & 15;
    const int jg = t >> 4;
    float acc = 0.f;
#pragma unroll
    for (int k = 0; k < 16; ++k) {
      const int j = (k << 4) + jg;
      acc += s_edges[j * E_ + e] * s_mask[j];
    }
    s_part[t] = acc;
  }
  __syncthreads();

  if (t < 16) {
    float s = 0.f;
#pragma unroll
    for (int jg = 0; jg < 16; ++jg) s += s_part[(jg << 4) + t];
    esum[(size_t)bid * E_ + t] = s;
  }
  if (t == 0) {
    float d = 0.f;
    for (int j = 0; j < N_; ++j) d += s_mask[j];
    deg[bid] = d;
  }
}

// ---------------------------------------------------------------------------
// K3: agg = mask @ mj  (per-batch 256x256 @ 256x64, WMMA K-loop of 64 steps)
//          + deg*(mi+bm) + esum @ W_edge   (fused epilogue)
// grid: 128 blocks (b, i-tile), 128 threads = 4 waves (one 16-col tile each).
// mask row-tile (16x256 = 16KB) staged in LDS.
// ---------------------------------------------------------------------------
__global__ void k3_aggregate(const float* __restrict__ mask,
                             const float* __restrict__ mj,
                             const float* __restrict__ mi,
                             const float* __restrict__ bm,
                             const float* __restrict__ deg,
                             const float* __restrict__ esum,
                             const float* __restrict__ Wm,
                             float* __restrict__ agg) {
  __shared__ float s_m[16 * N_];        // [i_local][j]

  const int blk  = blockIdx.x;
  const int b    = blk >> 4;
  const int i0   = (blk & 15) << 4;
  const int tid  = threadIdx.x;
  const int lane = tid & 31;
  const int w    = tid >> 5;
  const int n0   = w << 4;
  const int sel  = (lane < 16) ? 0 : 2;
  const int col  = lane & 15;
  const int roff = (lane < 16) ? 0 : 8;

  for (int idx = tid; idx < 16 * N_; idx += 128) {
    const int il = idx >> 8;
    const int j  = idx & (N_ - 1);
    s_m[idx] = mask[((size_t)(b * N_ + i0 + il)) * N_ + j];
  }
  __syncthreads();

  const float* mjb = mj + (size_t)b * N_ * M_;
  const float* arow = s_m + (size_t)col * N_;

  v8f c = {};
#pragma unroll 4
  for (int k = 0; k < N_; k += 4) {
    v2f a, bb;
    a.x  = arow[k + sel];
    a.y  = arow[k + 1 + sel];
    bb.x = mjb[(size_t)(k + sel) * M_ + n0 + col];
    bb.y = mjb[(size_t)(k + 1 + sel) * M_ + n0 + col];
    c = wmma4(a, bb, c);
  }

  const float* Wedge = Wm + 128 * M_;   // rows 128..143
#pragma unroll
  for (int r = 0; r < 8; ++r) {
    const int gr = b * N_ + i0 + r + roff;
    const int m  = n0 + col;
    float v = c[r];
    v += deg[gr] * (mi[(size_t)gr * M_ + m] + bm[m]);
    const float* es = esum + (size_t)gr * E_;
#pragma unroll
    for (int e = 0; e < E_; ++e) v += es[e] * Wedge[e * M_ + m];
    agg[(size_t)gr * M_ + m] = v;
  }
}

// ---------------------------------------------------------------------------
// K4: GRU.  xg = agg@W_gru + b0 (16x192), hg = nodes@U_gru + b1, then gates.
// grid: 128 blocks (b, i-tile), 256 threads = 8 waves; 24 WMMA output tiles
// -> 3 tiles per wave; results staged in LDS, elementwise fused at the end.
// ---------------------------------------------------------------------------
__global__ void k4_gru(const float* __restrict__ agg,
                       const float* __restrict__ nodes,
                       const float* __restrict__ W_gru,
                       const float* __restrict__ U_gru,
                       const float* __restrict__ b_gru,
                       float* __restrict__ out) {
  __shared__ float s_xg[16 * 192];
  __shared__ float s_hg[16 * 192];

  const int blk  = blockIdx.x;
  const int b    = blk >> 4;
  const int i0   = (blk & 15) << 4;
  const int tid  = threadIdx.x;
  const int lane = tid & 31;
  const int w    = tid >> 5;
  const int sel  = (lane < 16) ? 0 : 2;
  const int col  = lane & 15;
  const int roff = (lane < 16) ? 0 : 8;

#pragma unroll
  for (int q = 0; q < 3; ++q) {
    const int tile = w * 3 + q;         // 0..23
    const int mat  = tile / 12;         // 0 = xg, 1 = hg
    const int n0   = (tile % 12) << 4;
    const float* Arow = (mat ? nodes : agg) + (size_t)(b * N_ + i0 + col) * F_;
    const float* Bmat = mat ? U_gru : W_gru;   // 64 x 192 row-major

    v8f c = {};
#pragma unroll
    for (int k = 0; k < F_; k += 4) {
      v2f a, bb;
      a.x  = Arow[k + sel];
      a.y  = Arow[k + 1 + sel];
      bb.x = Bmat[(k + sel) * 192 + n0 + col];
      bb.y = Bmat[(k + 1 + sel) * 192 + n0 + col];
      c = wmma4(a, bb, c);
    }

    const float bias = b_gru[mat * 192 + n0 + col];
    float* dst = mat ? s_hg : s_xg;
#pragma unroll
    for (int r = 0; r < 8; ++r)
      dst[(r + roff) * 192 + n0 + col] = c[r] + bias;
  }
  __syncthreads();

  // gates: z = sig(xz+hz), r = sig(xr+hr), hh = tanh(xh + r*hc)
  for (int idx = tid; idx < 16 * F_; idx += 256) {
    const int il = idx >> 6;
    const int f  = idx & 63;
    const float xz = s_xg[il * 192 + f],       hz = s_hg[il * 192 + f];
    const float xr = s_xg[il * 192 + 64 + f],  hr = s_hg[il * 192 + 64 + f];
    const float xh = s_xg[il * 192 + 128 + f], hc = s_hg[il * 192 + 128 + f];
    const float z  = 1.f / (1.f + __expf(-(xz + hz)));
    const float r  = 1.f / (1.f + __expf(-(xr + hr)));
    const float hh = tanhf(xh + r * hc);
    const size_t gi = (size_t)(b * N_ + i0 + il) * F_ + f;
    const float h  = nodes[gi];
    out[gi] = z * h + (1.f - z) * hh;
  }
}

// ---------------------------------------------------------------------------
extern "C" void kernel_launch(void* const* d_in, const int* in_sizes, int n_in,
                              void* d_out, int out_size, void* d_ws, size_t ws_size,
                              hipStream_t stream) {
  const float* nodes = (const float*)d_in[0];   // (8,256,64)
  const float* edges = (const float*)d_in[1];   // (8,256,256,16)
  const float* Wm    = (const float*)d_in[2];   // (144,64)
  const float* bm    = (const float*)d_in[3];   // (64,)
  const float* W_gru = (const float*)d_in[4];   // (64,192)
  const float* U_gru = (const float*)d_in[5];   // (64,192)
  const float* b_gru = (const float*)d_in[6];   // (2,192)
  float* out = (float*)d_out;                   // (8,256,64)

  float* ws   = (float*)d_ws;
  float* mi   = ws;                 // 131072
  float* mj   = ws + 131072;        // 131072
  float* mask = ws + 262144;        // 524288
  float* deg  = ws + 786432;        // 2048
  float* esum = ws + 788480;        // 32768
  float* agg  = ws + 821248;        // 131072  (total ~3.8 MB)

  k1_project  <<<128, 256, 0, stream>>>(nodes, Wm, mi, mj);
  k2_mask_esum<<<B_ * N_, 256, 0, stream>>>(edges, mask, deg, esum);
  k3_aggregate<<<128, 128, 0, stream>>>(mask, mj, mi, bm, deg, esum, Wm, agg);
  k4_gru      <<<128, 256, 0, stream>>>(agg, nodes, W_gru, U_gru, b_gru, out);
}